// SAGE_8555574854331
// MI455X (gfx1250) — compile-verified
//
#include <hip/hip_runtime.h>

typedef __attribute__((ext_vector_type(2))) float v2f;
typedef __attribute__((ext_vector_type(8))) float v8f;

#define NN   40000
#define NE   640000
#define HIDF 128

// ---------------------------------------------------------------- utilities
__global__ void zero_f32_k(float* __restrict__ p, int n4) {
  int i = blockIdx.x * blockDim.x + threadIdx.x;
  if (i < n4) {
    float4 z; z.x = 0.f; z.y = 0.f; z.z = 0.f; z.w = 0.f;
    ((float4*)p)[i] = z;
  }
}

__global__ void deg_k(const int* __restrict__ dst, float* __restrict__ deg, int nE) {
  int e = blockIdx.x * blockDim.x + threadIdx.x;
  if (e < nE) atomicAdd(&deg[dst[e]], 1.0f);
}

__global__ void invdeg_k(float* __restrict__ deg, int n) {
  int i = blockIdx.x * blockDim.x + threadIdx.x;
  if (i < n) deg[i] = 1.0f / fmaxf(deg[i], 1.0f);
}

// Pack both weight matrices column-major in K-pairs so one b128 load feeds a
// whole WMMA k-step: packed[n*64 + kp] = {Ws[2kp][n], Ws[2kp+1][n],
//                                         Wn[2kp][n], Wn[2kp+1][n]}
// Padded to NCpad columns (zeros) so the GEMM hot loop needs no column guard.
__global__ void pack_w_k(const float* __restrict__ Ws,
                         const float* __restrict__ Wn,
                         float* __restrict__ packed, int NC, int total) {
  int i = blockIdx.x * blockDim.x + threadIdx.x;     // i = n*64 + kp
  if (i < total) {
    int n  = i >> 6;
    int kp = i & 63;
    float4 v; v.x = 0.f; v.y = 0.f; v.z = 0.f; v.w = 0.f;
    if (n < NC) {
      v.x = Ws[(2 * kp) * NC + n];
      v.y = Ws[(2 * kp + 1) * NC + n];
      v.z = Wn[(2 * kp) * NC + n];
      v.w = Wn[(2 * kp + 1) * NC + n];
    }
    ((float4*)packed)[i] = v;
  }
}

// Scatter-add: one thread owns 4 consecutive features of one edge.
// b128 gather of h[src], 4 float atomics into agg[dst] (L2-resident traffic).
__global__ void scatter_k(const float* __restrict__ h,
                          const int* __restrict__ src,
                          const int* __restrict__ dst,
                          float* __restrict__ agg, int nE) {
  int t = blockIdx.x * 256 + threadIdx.x;
  int e = t >> 5;                 // 32 threads per edge
  int c = (t & 31) << 2;          // 4 features per thread
  if (e < nE) {
    int s = src[e];
    int d = dst[e];
    float4 hv = *(const float4*)&h[s * HIDF + c];
    float* ap = &agg[d * HIDF + c];
    atomicAdd(ap + 0, hv.x);
    atomicAdd(ap + 1, hv.y);
    atomicAdd(ap + 2, hv.z);
    atomicAdd(ap + 3, hv.w);
  }
}

// ---------------------------------------------------------------- fused SAGE GEMM
// out[16 rows x NC cols] = h*Wself + (agg*invdeg)*Wneigh + bias (opt ReLU)
// 256 threads = 8 waves; block owns 16 rows, wave w owns N-tile w.
// WMMA f32 16x16x4; per k-step: 2x ds_load b64 (A) + 1x global b128 (packed B)
// + 2x v_wmma. fp32 end to end (GEMM is compute-trivial; precision is free).
template <int NC, bool RELU>
__global__ __launch_bounds__(256, 2)
void sage_gemm_k(const float* __restrict__ h,
                 const float* __restrict__ agg,
                 const float* __restrict__ invdeg,
                 const float* __restrict__ Wpacked,   // padded to NT*16 cols
                 const float* __restrict__ bias,
                 float* __restrict__ out) {
  constexpr int NT = (NC + 15) / 16;          // N tiles (8 for 128, 3 for 47)
  __shared__ float sSelf[16][HIDF + 4];       // pad keeps b64 align, kills conflicts
  __shared__ float sNeigh[16][HIDF + 4];

  const int tid = threadIdx.x;
  const int rowBase = blockIdx.x * 16;        // 40000/16 = 2500 blocks, exact

  // Stage both A tiles; fold mean-normalization into the load.
  for (int i = tid; i < 16 * HIDF; i += 256) {
    int r = i >> 7;
    int c = i & (HIDF - 1);
    int g = rowBase + r;
    sSelf[r][c]  = h[g * HIDF + c];
    sNeigh[r][c] = agg[g * HIDF + c] * invdeg[g];
  }
  __syncthreads();

  const int wave = tid >> 5;
  const int lane = tid & 31;
  if (wave < NT) {                            // wave-uniform: EXEC all-ones inside
    const int m    = lane & 15;               // A row / D column index
    const int kOff = (lane >> 4) << 1;        // lanes 16-31 hold K+2, K+3
    const int nIdx = wave * 16 + m;           // padded column: always in-bounds
    const float4* __restrict__ Wp = (const float4*)Wpacked;

    v8f acc = {};
#pragma unroll
    for (int k0 = 0; k0 < HIDF; k0 += 4) {
      // A 16x4 f32 layout: VGPR0=K(k0+kOff), VGPR1=K(k0+kOff+1)
      v2f aS = *(const v2f*)&sSelf[m][k0 + kOff];
      v2f aN = *(const v2f*)&sNeigh[m][k0 + kOff];
      // one b128 = {bSelf.k, bSelf.k+1, bNeigh.k, bNeigh.k+1} for this lane
      float4 bw = Wp[nIdx * 64 + ((k0 + kOff) >> 1)];
      v2f bS; bS.x = bw.x; bS.y = bw.y;
      v2f bN; bN.x = bw.z; bN.y = bw.w;

      acc = __builtin_amdgcn_wmma_f32_16x16x4_f32(false, aS, false, bS,
                                                  (short)0, acc, false, false);
      acc = __builtin_amdgcn_wmma_f32_16x16x4_f32(false, aN, false, bN,
                                                  (short)0, acc, false, false);
    }

    // C/D 16x16 f32 layout: VGPR r -> row r (lanes 0-15) / r+8 (lanes 16-31)
    const bool colOK = (nIdx < NC);
    const float bv = colOK ? bias[nIdx] : 0.f;
    const int rowHalf = (lane >> 4) << 3;
#pragma unroll
    for (int r = 0; r < 8; ++r) {
      if (colOK) {
        float v = acc[r] + bv;
        if (RELU) v = fmaxf(v, 0.f);
        out[(rowBase + r + rowHalf) * NC + nIdx] = v;
      }
    }
  }
}

// ---------------------------------------------------------------- launch
extern "C" void kernel_launch(void* const* d_in, const int* in_sizes, int n_in,
                              void* d_out, int out_size, void* d_ws, size_t ws_size,
                              hipStream_t stream) {
  (void)in_sizes; (void)n_in; (void)out_size; (void)ws_size;

  const float* x   = (const float*)d_in[0];
  const int*   src = (const int*)d_in[1];
  const int*   dst = (const int*)d_in[2];
  const float* Ws0 = (const float*)d_in[3];
  const float* Wn0 = (const float*)d_in[4];
  const float* b0  = (const float*)d_in[5];
  const float* Ws1 = (const float*)d_in[6];
  const float* Wn1 = (const float*)d_in[7];
  const float* b1  = (const float*)d_in[8];
  const float* Ws2 = (const float*)d_in[9];
  const float* Wn2 = (const float*)d_in[10];
  const float* b2  = (const float*)d_in[11];
  float* out = (float*)d_out;

  // workspace: invdeg | h0 | agg | h1 | packed weights   (~62 MB)
  float* ws     = (float*)d_ws;
  float* invdeg = ws;                         // 40960 floats (padded)
  float* bufH   = invdeg + 40960;             // NN*HIDF
  float* bufAgg = bufH + NN * HIDF;           // NN*HIDF
  float* bufH2  = bufAgg + NN * HIDF;         // NN*HIDF
  float* packW0 = bufH2 + NN * HIDF;          // 128*64*4
  float* packW1 = packW0 + 128 * 64 * 4;      // 128*64*4
  float* packW2 = packW1 + 128 * 64 * 4;      // 48*64*4 (padded to 3 tiles)

  const int nodeBlocks    = (NN + 255) / 256;
  const int edgeBlocks    = (NE + 255) / 256;
  const int aggN4         = NN * HIDF / 4;
  const int aggZeroBlocks = (aggN4 + 255) / 256;
  const int scatterBlocks = (NE * 32 + 255) / 256;   // 4 feats/thread
  const int gemmBlocks    = NN / 16;                 // 2500, exact

  // one-time weight transpose/pack (tiny; reused 2500x from L2)
  pack_w_k<<<(128 * 64 + 255) / 256, 256, 0, stream>>>(Ws0, Wn0, packW0, 128, 128 * 64);
  pack_w_k<<<(128 * 64 + 255) / 256, 256, 0, stream>>>(Ws1, Wn1, packW1, 128, 128 * 64);
  pack_w_k<<<(48 * 64 + 255) / 256, 256, 0, stream>>>(Ws2, Wn2, packW2, 47, 48 * 64);

  // inverse degree (shared by all layers)
  zero_f32_k<<<(NN / 4 + 255) / 256, 256, 0, stream>>>(invdeg, NN / 4);
  deg_k<<<edgeBlocks, 256, 0, stream>>>(dst, invdeg, NE);
  invdeg_k<<<nodeBlocks, 256, 0, stream>>>(invdeg, NN);

  // layer 0
  zero_f32_k<<<aggZeroBlocks, 256, 0, stream>>>(bufAgg, aggN4);
  scatter_k<<<scatterBlocks, 256, 0, stream>>>(x, src, dst, bufAgg, NE);
  sage_gemm_k<HIDF, true><<<gemmBlocks, 256, 0, stream>>>(
      x, bufAgg, invdeg, packW0, b0, bufH);

  // layer 1
  zero_f32_k<<<aggZeroBlocks, 256, 0, stream>>>(bufAgg, aggN4);
  scatter_k<<<scatterBlocks, 256, 0, stream>>>(bufH, src, dst, bufAgg, NE);
  sage_gemm_k<HIDF, true><<<gemmBlocks, 256, 0, stream>>>(
      bufH, bufAgg, invdeg, packW1, b1, bufH2);

  // layer 2 (47 output cols, no ReLU) -> d_out
  zero_f32_k<<<aggZeroBlocks, 256, 0, stream>>>(bufAgg, aggN4);
  scatter_k<<<scatterBlocks, 256, 0, stream>>>(bufH2, src, dst, bufAgg, NE);
  sage_gemm_k<47, false><<<gemmBlocks, 256, 0, stream>>>(
      bufH2, bufAgg, invdeg, packW2, b2, out);
}